// LightGCN_84705345012080
// MI455X (gfx1250) — compile-verified
//
#include <hip/hip_runtime.h>
#include <stdint.h>

// ---- Problem constants (from reference) ----
#define NU     300000
#define NI     200000
#define NNODE  (NU + NI)          // 500000
#define DIM    64
#define NEDGE  2500000
#define TPB    256
#define SPMM_WAVES (TPB / 32)     // 8 waves per block (wave32)

typedef float v4f __attribute__((ext_vector_type(4)));

// ---------------------------------------------------------------------------
// d_out layout (floats): [users_final | users_copy | items_final | items_copy]
// accBase(n): node n's accumulator (final) row in d_out.
// cpBase(n):  node n's "copy" row in d_out (borrowed as ping-pong x-buffer
//             during the layer loop, rewritten with input copies at the end).
// ---------------------------------------------------------------------------
__device__ __forceinline__ size_t accBase(int n) {
  return (size_t)(n < NU ? n : NU + n) * DIM;
}
__device__ __forceinline__ size_t cpBase(int n) {
  return (size_t)(n < NU ? NU + n : NU + NI + n) * DIM;
}
__device__ __forceinline__ size_t xBase(int n, int mapped) {
  return mapped ? cpBase(n) : (size_t)n * DIM;
}

// ---- CDNA5 async gather: one 16B chunk per lane, DMA'd into LDS ------------
__device__ __forceinline__ void asyncGather16(const float* gsrc, unsigned ldsDstByte) {
  // GLOBAL_LOAD_ASYNC_TO_LDS_B128: VDST = LDS byte-offset VGPR, VADDR = 64-bit
  // global address. Tracked by ASYNCcnt; async loads complete in order.
  asm volatile("global_load_async_to_lds_b128 %0, %1, off"
               :: "v"(ldsDstByte), "v"(gsrc)
               : "memory");
}
__device__ __forceinline__ void waitAsyncLe1() {
  asm volatile("s_wait_asynccnt 0x1" ::: "memory");
}

// Confirmed from disasm: lowers to a single no-return hardware
// global_atomic_add_f32 ... scope:SCOPE_DEV (no CAS loop).
__device__ __forceinline__ void atomAddF32(float* p, float v) {
  (void)__hip_atomic_fetch_add(p, v, __ATOMIC_RELAXED, __HIP_MEMORY_SCOPE_AGENT);
}

// ---------------------------------------------------------------------------
// K1: degree count via hardware f32 atomics; edge stream NT.
// ---------------------------------------------------------------------------
__global__ void __launch_bounds__(TPB)
k_deg(const int* __restrict__ col, float* __restrict__ deg) {
  int e = blockIdx.x * blockDim.x + threadIdx.x;
  if (e >= NEDGE) return;
  int c = __builtin_nontemporal_load(col + e);
  atomAddF32(deg + c, 1.0f);
}

// ---------------------------------------------------------------------------
// SpMM:  xout[col] += norm * xin[row]
// Each wave handles 2 edges/iter: lanes 0-15 -> edge 2p, lanes 16-31 -> 2p+1;
// each lane owns 4 dims. The 2x64-float row gather for the NEXT iteration is
// one async-DMA-to-LDS instruction per wave (double-buffered), hiding random
// gather latency behind the current iteration's global f32 atomics.
// ---------------------------------------------------------------------------
__device__ __forceinline__ void issueGather(const float* __restrict__ xin, int inMapped,
                                            const int* __restrict__ row,
                                            int p, int sub, int l16, unsigned ldsDst) {
  int e = 2 * p + sub;                        // NEDGE even -> valid for p < nPairs
  int r = __builtin_nontemporal_load(row + e);
  const float* g = xin + xBase(r, inMapped) + (size_t)l16 * 4;
  asyncGather16(g, ldsDst);
}

__global__ void __launch_bounds__(TPB)
k_spmm(const float* __restrict__ xin, int inMapped,
       float* __restrict__ xout, int outMapped,
       const int* __restrict__ row, const int* __restrict__ col,
       const float* __restrict__ nrm, int nPairs) {
  __shared__ float lds[SPMM_WAVES][2][2 * DIM];   // 8KB: per-wave double buffer
  const int lane = threadIdx.x & 31;
  const int wv   = threadIdx.x >> 5;
  const int sub  = lane >> 4;      // which edge of the pair
  const int l16  = lane & 15;      // which float4 of the row
  const int waveG      = blockIdx.x * SPMM_WAVES + wv;
  const int waveStride = gridDim.x * SPMM_WAVES;
  if (waveG >= nPairs) return;

  unsigned ldsDst[2];
  ldsDst[0] = (unsigned)(uintptr_t)(&lds[wv][0][0]) + (unsigned)lane * 16u;
  ldsDst[1] = (unsigned)(uintptr_t)(&lds[wv][1][0]) + (unsigned)lane * 16u;

  int sel = 0;
  issueGather(xin, inMapped, row, waveG, sub, l16, ldsDst[0]);   // prologue

  for (int p = waveG; p < nPairs; p += waveStride, sel ^= 1) {
    int pn = p + waveStride;
    int pIss = (pn < nPairs) ? pn : p;       // harmless dummy re-issue at tail
    issueGather(xin, inMapped, row, pIss, sub, l16, ldsDst[sel ^ 1]);
    waitAsyncLe1();                          // buffer `sel` is now complete

    int e = 2 * p + sub;
    float nv = __builtin_nontemporal_load(nrm + e);
    int   c  = __builtin_nontemporal_load(col + e);
    v4f v = *(const v4f*)&lds[wv][sel][lane * 4];
    float* o = xout + xBase(c, outMapped) + (size_t)l16 * 4;
    atomAddF32(o + 0, nv * v.x);
    atomAddF32(o + 1, nv * v.y);
    atomAddF32(o + 2, nv * v.z);
    atomAddF32(o + 3, nv * v.w);
  }
  // <=1 dangling async at exit; s_endpgm performs implicit wait-idle.
}

// ---------------------------------------------------------------------------
// K0: xA = x0 = concat(users, items); acc init = x0; zero layer-0 target
// (d_out copy region) and the degree array.  (N*16 float4 threads)
// ---------------------------------------------------------------------------
__global__ void __launch_bounds__(TPB)
k_init(const float* __restrict__ users, const float* __restrict__ items,
       float* __restrict__ xA, float* __restrict__ out, float* __restrict__ deg) {
  int idx = blockIdx.x * blockDim.x + threadIdx.x;
  if (idx >= NNODE * 16) return;
  int n = idx >> 4, q = idx & 15;
  const v4f v = (n < NU)
      ? ((const v4f*)users)[(size_t)n * 16 + q]
      : ((const v4f*)items)[(size_t)(n - NU) * 16 + q];
  ((v4f*)xA)[(size_t)n * 16 + q] = v;
  *(v4f*)(out + accBase(n) + (size_t)q * 4) = v;          // acc := x0
  v4f z = {0.f, 0.f, 0.f, 0.f};
  *(v4f*)(out + cpBase(n) + (size_t)q * 4) = z;           // zero layer-0 target
  if (q == 0) deg[n] = 0.0f;
}

// K2: deg_inv_sqrt
__global__ void __launch_bounds__(TPB)
k_dis(const float* __restrict__ deg, float* __restrict__ dis) {
  int n = blockIdx.x * blockDim.x + threadIdx.x;
  if (n >= NNODE) return;
  float d = deg[n];
  dis[n] = (d > 0.0f) ? rsqrtf(fmaxf(d, 1.0f)) : 0.0f;
}

// K3: per-edge norm = dis[row]*dis[col]
__global__ void __launch_bounds__(TPB)
k_norm(const int* __restrict__ row, const int* __restrict__ col,
       const float* __restrict__ dis, float* __restrict__ nrm) {
  int e = blockIdx.x * blockDim.x + threadIdx.x;
  if (e >= NEDGE) return;
  int r = __builtin_nontemporal_load(row + e);
  int c = __builtin_nontemporal_load(col + e);
  nrm[e] = dis[r] * dis[c];
}

// K4 (fused): acc += layer output; zero the NEXT layer's target buffer
// (= the buffer the SpMM just finished reading).
__global__ void __launch_bounds__(TPB)
k_accum_zero(float* __restrict__ out, const float* __restrict__ src, int srcMapped,
             float* __restrict__ nxt, int nxtMapped) {
  int idx = blockIdx.x * blockDim.x + threadIdx.x;
  if (idx >= NNODE * 16) return;
  int n = idx >> 4, q = idx & 15;
  v4f* a = (v4f*)(out + accBase(n) + (size_t)q * 4);
  const v4f s = *(const v4f*)(src + xBase(n, srcMapped) + (size_t)q * 4);
  v4f t = *a;
  t += s;
  *a = t;
  v4f z = {0.f, 0.f, 0.f, 0.f};
  *(v4f*)(nxt + xBase(n, nxtMapped) + (size_t)q * 4) = z;
}

// K5: final = (acc + x4) / 25 (NT store); rewrite borrowed copy regions with
// the input copies (NT store — outputs are never re-read on device).
__global__ void __launch_bounds__(TPB)
k_final(float* __restrict__ out, const float* __restrict__ users,
        const float* __restrict__ items, const float* __restrict__ x4) {
  int idx = blockIdx.x * blockDim.x + threadIdx.x;
  if (idx >= NNODE * 16) return;
  int n = idx >> 4, q = idx & 15;
  const float s = 1.0f / 25.0f;   // 1/(L+1) * mean over (L+1)
  v4f t = *(const v4f*)(out + accBase(n) + (size_t)q * 4);
  const v4f l4 = ((const v4f*)x4)[(size_t)n * 16 + q];
  t = (t + l4) * s;
  __builtin_nontemporal_store(t, (v4f*)(out + accBase(n) + (size_t)q * 4));
  const v4f v = (n < NU)
      ? ((const v4f*)users)[(size_t)n * 16 + q]
      : ((const v4f*)items)[(size_t)(n - NU) * 16 + q];
  __builtin_nontemporal_store(v, (v4f*)(out + cpBase(n) + (size_t)q * 4));
}

// ---------------------------------------------------------------------------
extern "C" void kernel_launch(void* const* d_in, const int* in_sizes, int n_in,
                              void* d_out, int out_size, void* d_ws, size_t ws_size,
                              hipStream_t stream) {
  (void)in_sizes; (void)n_in; (void)out_size; (void)ws_size;
  const int*   edge  = (const int*)d_in[0];
  const int*   row   = edge;            // edge_index[0]
  const int*   col   = edge + NEDGE;    // edge_index[1]
  const float* users = (const float*)d_in[1];
  const float* items = (const float*)d_in[2];
  float*       out   = (float*)d_out;

  // Workspace: deg | dis | norm | xA   (~142 MB total)
  float* deg = (float*)d_ws;
  float* dis = deg + NNODE;
  float* nrm = dis + NNODE;
  float* xA  = nrm + NEDGE;

  const int gVec = (NNODE * 16 + TPB - 1) / TPB;
  const int gE   = (NEDGE + TPB - 1) / TPB;
  const int gN   = (NNODE + TPB - 1) / TPB;

  k_init<<<gVec, TPB, 0, stream>>>(users, items, xA, out, deg);
  k_deg <<<gE,   TPB, 0, stream>>>(col, deg);
  k_dis <<<gN,   TPB, 0, stream>>>(deg, dis);
  k_norm<<<gE,   TPB, 0, stream>>>(row, col, dis, nrm);

  const int nPairs = NEDGE / 2;
  const int spmmBlocks = 4096;            // 32768 waves, ~38 pair-iters each
  // Layer l even: xA(contig) -> d_out copy region (mapped); odd: reverse.
  for (int l = 0; l < 4; ++l) {
    const int outMapped = (l % 2 == 0) ? 1 : 0;
    const float* xi = outMapped ? (const float*)xA : (const float*)out;
    const int inMapped = outMapped ? 0 : 1;
    float* xo = outMapped ? out : xA;
    k_spmm<<<spmmBlocks, TPB, 0, stream>>>(xi, inMapped, xo, outMapped,
                                           row, col, nrm, nPairs);
    if (l < 3) {
      // next target is the buffer we just read (xi); zero it while accumulating
      k_accum_zero<<<gVec, TPB, 0, stream>>>(out, xo, outMapped,
                                             (float*)(outMapped ? (void*)xA : (void*)out),
                                             outMapped ? 0 : 1);
    }
  }
  // x4 lives in xA after layer 3 (outMapped==0)
  k_final<<<gVec, TPB, 0, stream>>>(out, users, items, xA);
}